// Seq2SeqLSTM_75926431858950
// MI455X (gfx1250) — compile-verified
//
#include <hip/hip_runtime.h>

// ---------------------------------------------------------------------------
// Seq2Seq LSTM (MI455X / gfx1250).
//   * decoder scan carry is seeded with ZEROS -> encoder output is dead code.
//   * decoder has no per-batch input -> output is identical across batch.
// => compute one 128-step, 4-layer LSTM decoder chain (GEMV recurrence,
//    K = 1024 for concat [x;h], N = 2048 gate outputs per layer) with exact
//    fp32 WMMA (v_wmma_f32_16x16x4_f32), then broadcast [128,512] -> d_out.
// A-fragment trick: D rows 1..15 are discarded, so every lane half just
// broadcasts the same xh pair -> no lane masking, no divergence.
// ---------------------------------------------------------------------------

typedef __attribute__((ext_vector_type(2))) float v2f;
typedef __attribute__((ext_vector_type(4))) float v4f;
typedef __attribute__((ext_vector_type(8))) float v8f;

#define SHID 512
#define FLEN 128
#define NLAY 4
#define BATCH 256

__device__ __forceinline__ float sigmoidf_(float x) {
  return 1.0f / (1.0f + expf(-x));
}

__global__ void lstm_init_state(float* __restrict__ hstate,
                                float* __restrict__ cstate) {
  int i = blockIdx.x * blockDim.x + threadIdx.x;
  if (i < NLAY * SHID) {
    hstate[i] = 0.0f;
    cstate[i] = 0.0f;
  }
}

// Reduce one 512-deep half of the GEMV with a depth-1 software pipeline.
//   p      : per-lane weight pointer, element [kk][nn] of this gate block (kk = 0 or 2 by lane half)
//            (b(k).x = p[k*512], b(k).y = p[k*512 + 512])
//   xv     : LDS vector source; lane half L loads xv[k0 + aoff .. +1]
__device__ __forceinline__ v8f gemv_half_512(const float* __restrict__ p,
                                             const float* xv, int aoff,
                                             v8f acc) {
  v2f a_c = *(const v2f*)(xv + aoff);  // k0 = 0
  v2f b_c;
  b_c.x = p[0];
  b_c.y = p[SHID];
#pragma unroll 8
  for (int k0 = 0; k0 < SHID - 4; k0 += 4) {
    // Prefetch 32 K-rows (64 KB) ahead; speculative -> safe past the block end.
    __builtin_prefetch(p + (size_t)(k0 + 32) * SHID, 0, 3);
    const size_t o = (size_t)(k0 + 4) * SHID;
    v2f a_n = *(const v2f*)(xv + (k0 + 4) + aoff);
    v2f b_n;
    b_n.x = p[o];
    b_n.y = p[o + SHID];
    acc = __builtin_amdgcn_wmma_f32_16x16x4_f32(false, a_c, false, b_c,
                                                (short)0, acc, false, false);
    a_c = a_n;
    b_c = b_n;
  }
  acc = __builtin_amdgcn_wmma_f32_16x16x4_f32(false, a_c, false, b_c, (short)0,
                                              acc, false, false);
  return acc;
}

// One decoder layer-step. Grid: 16 blocks x 256 threads (8 wave32).
// Block wg owns hidden columns [wg*32, wg*32+32); wave j computes gate
// g = j>>1, N-tile n0 = wg*32 + (j&1)*16, reducing over K = 1024 ([x;h]).
__global__ __launch_bounds__(256) void lstm_decoder_layer_step(
    const float* __restrict__ Wx,    // [4][4][512][512] dec_Wx
    const float* __restrict__ Wh,    // [4][4][512][512] dec_Wh
    const float* __restrict__ bias,  // [4][4][512]      dec_b
    float* __restrict__ hstate,      // [4][512]
    float* __restrict__ cstate,      // [4][512]
    float* __restrict__ hseq,        // [128][512]
    int l, int t) {
  __shared__ float xh[2 * SHID];    // concat [x ; h] for this layer
  __shared__ float ldspre[4 * 32];  // 4 gates x 32 columns

  const int tid = threadIdx.x;
  const int wg = blockIdx.x;  // 0..15

  // Stage xh into LDS. Layer 0's x is previous step's top-layer h
  // (hstate[3]); at t==0 it is the zero-initialized state (matches state0=0).
  const float* xsrc = (l == 0) ? (hstate + 3 * SHID) : (hstate + (l - 1) * SHID);
  const float* hsrc = hstate + l * SHID;
  for (int i = tid; i < SHID; i += 256) {
    xh[i] = xsrc[i];
    xh[SHID + i] = hsrc[i];
  }
  __syncthreads();

  const int wave = tid >> 5;
  const int lane = tid & 31;
  const int g = wave >> 1;                   // gate index 0..3 (f,i,g,o)
  const int n0 = wg * 32 + (wave & 1) * 16;  // column tile base within gate

  // A fragment layout (16x4 fp32): lanes 0-15 carry K=k0,k0+1; lanes 16-31
  // carry K=k0+2,k0+3. All lanes in a half load the SAME pair (rows 1..15 of
  // D just replicate row 0 and are ignored) -> broadcast ds_load_b64.
  const int aoff = (lane >> 4) << 1;  // 0 or 2
  // B fragment: lane half selects K-pair, N = n0 + (lane & 15).
  const int kk = (lane < 16) ? 0 : 2;
  const int nn = n0 + (lane & 15);

  const size_t gsz = (size_t)SHID * SHID;
  const float* pWx = Wx + (size_t)(l * 4 + g) * gsz + (size_t)kk * SHID + nn;
  const float* pWh = Wh + (size_t)(l * 4 + g) * gsz + (size_t)kk * SHID + nn;

  v8f acc = {};
  acc = gemv_half_512(pWx, xh, aoff, acc);         // x . Wx
  acc = gemv_half_512(pWh, xh + SHID, aoff, acc);  // h . Wh

  // Row M=0 of D lives in acc[0] on lanes 0..15 (N = n0 + lane).
  if (lane < 16) {
    const int col = n0 + lane;  // column within [0,512)
    const float pre = acc[0] + bias[(l * 4 + g) * SHID + col];
    ldspre[g * 32 + (col - wg * 32)] = pre;
  }
  __syncthreads();

  // Cell update for this block's 32 hidden columns (gate order f,i,g,o).
  if (tid < 32) {
    const int col = wg * 32 + tid;
    const float f = sigmoidf_(ldspre[0 * 32 + tid]);
    const float i = sigmoidf_(ldspre[1 * 32 + tid]);
    const float gg = tanhf(ldspre[2 * 32 + tid]);
    const float o = sigmoidf_(ldspre[3 * 32 + tid]);
    const float c = f * cstate[l * SHID + col] + i * gg;
    const float h = o * tanhf(c);
    cstate[l * SHID + col] = c;
    hstate[l * SHID + col] = h;
    if (l == NLAY - 1) hseq[t * SHID + col] = h;
  }
}

// Broadcast hseq[128][512] across batch into out[256][128][512].
// 64 MB streaming write -> nontemporal 128-bit stores; hseq is L2-resident.
__global__ __launch_bounds__(256) void lstm_broadcast_out(
    const v4f* __restrict__ hseq, v4f* __restrict__ out, int total4) {
  int idx = blockIdx.x * blockDim.x + threadIdx.x;
  if (idx >= total4) return;
  const int sf = idx & 127;           // float4 index within hidden row
  const int trow = (idx >> 7) & 127;  // time step
  v4f v = hseq[trow * (SHID / 4) + sf];
  __builtin_nontemporal_store(v, &out[idx]);
}

extern "C" void kernel_launch(void* const* d_in, const int* in_sizes, int n_in,
                              void* d_out, int out_size, void* d_ws,
                              size_t ws_size, hipStream_t stream) {
  // setup_inputs order: x, enc0_Wx, enc0_Wh, enc0_b, encr_Wx, encr_Wh,
  // encr_b, dec_Wx, dec_Wh, dec_b, fl.  Encoder inputs are dead code.
  const float* dec_Wx = (const float*)d_in[7];
  const float* dec_Wh = (const float*)d_in[8];
  const float* dec_b = (const float*)d_in[9];

  float* hstate = (float*)d_ws;          // 4*512 floats
  float* cstate = hstate + NLAY * SHID;  // 4*512 floats
  float* hseq = cstate + NLAY * SHID;    // 128*512 floats

  lstm_init_state<<<(NLAY * SHID + 255) / 256, 256, 0, stream>>>(hstate,
                                                                 cstate);

  for (int t = 0; t < FLEN; ++t) {
    for (int l = 0; l < NLAY; ++l) {
      lstm_decoder_layer_step<<<16, 256, 0, stream>>>(dec_Wx, dec_Wh, dec_b,
                                                      hstate, cstate, hseq, l,
                                                      t);
    }
  }

  const int total4 = BATCH * FLEN * SHID / 4;
  lstm_broadcast_out<<<(total4 + 255) / 256, 256, 0, stream>>>(
      (const v4f*)hseq, (v4f*)d_out, total4);
}